// SConv3d_24266565222406
// MI455X (gfx1250) — compile-verified
//
#include <hip/hip_runtime.h>

typedef __attribute__((ext_vector_type(2))) float v2f;
typedef __attribute__((ext_vector_type(4))) float v4f;
typedef __attribute__((ext_vector_type(8))) float v8f;

#define INC  16
#define OUTC 16
#define KOFF 27
#define LDS_ROW 18                       // padded row stride (floats): kills LDS bank conflicts
#define WBLK   (INC * LDS_ROW)           // 288 floats per padded 16x16 weight matrix
#define NW     (KOFF + 1)                // 27 conv matrices + 1 point matrix

__global__ __launch_bounds__(256)
void sconv3d_wmma_f32(const float* __restrict__ feats,
                      const float* __restrict__ Wc,
                      const float* __restrict__ Wp,
                      const float* __restrict__ bp,
                      const int*   __restrict__ nidx,
                      const int*   __restrict__ nmask,
                      float*       __restrict__ out,
                      int n, int numTiles)
{
    // LDS: padded weights + bias  (28*288 + 16 floats ~= 31.6 KB of 320 KB/WGP)
    __shared__ float sW[NW * WBLK + OUTC];

    const int tid = threadIdx.x;
    // element (w, r, c) -> sW[w*WBLK + r*LDS_ROW + c]
    for (int i = tid; i < KOFF * 256; i += 256) {
        const int w = i >> 8, rc = i & 255;
        sW[w * WBLK + (rc >> 4) * LDS_ROW + (rc & 15)] = Wc[i];
    }
    {
        const int r = tid >> 4, c = tid & 15;     // 256 threads == 256 elements
        sW[KOFF * WBLK + r * LDS_ROW + c] = Wp[tid];
    }
    if (tid < OUTC) sW[NW * WBLK + tid] = bp[tid];
    __syncthreads();

    const int wave = tid >> 5;
    const int lane = tid & 31;
    const int tile = blockIdx.x * 8 + wave;
    if (tile >= numTiles) return;                 // wave-uniform: EXEC all-1s for WMMA

    const int  mrow = lane & 15;                  // M row (and B column) this lane serves
    const bool hi   = lane >= 16;
    const int  c0   = hi ? 8 : 0;                 // this lane's contiguous channel window
    const int  base = tile * 16;
    int row = base + mrow;
    if (row >= n) row = n - 1;

    // Dual accumulators: halves the WMMA RAW-dependency chain.
    // D/C layout: VGPR r -> (M = r + (hi?8:0), N = lane&15); bias broadcast per column
    const float bias = sW[NW * WBLK + mrow];
    v8f accA, accB;
    #pragma unroll
    for (int i = 0; i < 8; ++i) { accA[i] = bias; accB[i] = 0.0f; }

    // One 16x16x16 f32 tile-matmul = 4 x 16x16x4 WMMAs, split across two acc chains.
    // K-permutation: K-slot (4j, 4j+1) [lanes 0-15] -> channels (2j, 2j+1);
    //                K-slot (4j+2,4j+3) [lanes16-31] -> channels (8+2j, 8+2j+1).
    // A fragment per lane = channels [c0, c0+8) == two b128 loads.
    #define TILE_MMA(a0, a1, wbase)                                              \
        {                                                                        \
            _Pragma("unroll")                                                    \
            for (int j = 0; j < 4; ++j) {                                        \
                v2f a;                                                           \
                if (j == 0)      { a.x = (a0).x; a.y = (a0).y; }                 \
                else if (j == 1) { a.x = (a0).z; a.y = (a0).w; }                 \
                else if (j == 2) { a.x = (a1).x; a.y = (a1).y; }                 \
                else             { a.x = (a1).z; a.y = (a1).w; }                 \
                v2f b;                                                           \
                b.x = (wbase)[(c0 + 2 * j    ) * LDS_ROW + mrow];                \
                b.y = (wbase)[(c0 + 2 * j + 1) * LDS_ROW + mrow];                \
                if (j < 2)                                                       \
                    accA = __builtin_amdgcn_wmma_f32_16x16x4_f32(                \
                              false, a, false, b, (short)0, accA, false, false); \
                else                                                             \
                    accB = __builtin_amdgcn_wmma_f32_16x16x4_f32(                \
                              false, a, false, b, (short)0, accB, false, false); \
            }                                                                    \
        }

    // ---- fused residual: acc += feats[tile rows] @ W_point + b ----
    {
        const float* frow = feats + (size_t)row * INC + c0;
        v4f a0 = *(const v4f*)(frow);
        v4f a1 = *(const v4f*)(frow + 4);
        const float* wp = &sW[KOFF * WBLK];
        TILE_MMA(a0, a1, wp);
    }

    // ---- 27 offsets: unconditional gather (L2-resident) * mask -> WMMA ----
    // Pointer-bump addressing: k -> k+1 slice is a uniform += n (no per-iter muls).
    const int* ip = nidx  + row;
    const int* mp = nmask + row;
    int gi = *ip;                                 // k = 0, software-pipelined
    int gm = *mp;
    #pragma unroll 3
    for (int k = 0; k < KOFF; ++k) {
        const int cgi = gi, cgm = gm;
        const float* grow = feats + (size_t)cgi * INC + c0;
        v4f a0 = *(const v4f*)(grow);             // global_load_b128
        v4f a1 = *(const v4f*)(grow + 4);         // global_load_b128
        if (k + 1 < KOFF) {                       // prefetch next k's index/mask
            ip += n; mp += n;
            gi = *ip; gm = *mp;
        }
        const float fm = (float)cgm;              // reference semantics: feats[idx] * mask
        a0 *= fm;
        a1 *= fm;
        const float* wk = &sW[k * WBLK];
        TILE_MMA(a0, a1, wk);
    }
    #undef TILE_MMA

    const v8f acc = accA + accB;

    // ---- store: VGPR r -> out[base + r + (hi?8:0)][mrow] ----
    const int rbase = base + (hi ? 8 : 0);
    float* orow = out + (size_t)rbase * OUTC + mrow;
    if (rbase + 7 < n) {
        #pragma unroll
        for (int r = 0; r < 8; ++r) orow[r * OUTC] = acc[r];
    } else {
        #pragma unroll
        for (int r = 0; r < 8; ++r)
            if (rbase + r < n) orow[r * OUTC] = acc[r];
    }
}

extern "C" void kernel_launch(void* const* d_in, const int* in_sizes, int n_in,
                              void* d_out, int out_size, void* d_ws, size_t ws_size,
                              hipStream_t stream) {
    const float* feats = (const float*)d_in[0];
    const float* Wc    = (const float*)d_in[1];
    const float* Wp    = (const float*)d_in[2];
    const float* bp    = (const float*)d_in[3];
    const int*   nidx  = (const int*)  d_in[4];
    const int*   nmask = (const int*)  d_in[5];
    float* out = (float*)d_out;

    const int n        = in_sizes[0] / INC;       // number of voxels
    const int numTiles = (n + 15) / 16;
    const int blocks   = (numTiles + 7) / 8;      // 8 waves (tiles) per 256-thread block

    sconv3d_wmma_f32<<<blocks, 256, 0, stream>>>(feats, Wc, Wp, bp, nidx, nmask,
                                                 out, n, numTiles);
}